// DenseNextBlk_41497974014440
// MI455X (gfx1250) — compile-verified
//
#include <hip/hip_runtime.h>
#include <stdint.h>

// ---------------------------------------------------------------------------
// DenseNextBlk on MI455X (gfx1250)
//
// Memory-bound gather + tiny conv (4 branches x [3x3, 48ch -> 1ch] + BN+ReLU).
// Roofline: ~100MB unique reads @ 23.3 TB/s ~= 4.5us, 0.45 GFLOP -> firmly
// memory-bound; working set fits the 192MB L2 so halo re-reads are cheap.
//
//   * One WG = one 16x16 output tile x one batch, all 4 branches together:
//     branch i reads channel i+32c', so branches 0..3 share 16B-contiguous
//     channel chunks in all three inputs.
//   * Each 16B chunk is moved with one global_load_async_to_lds_b128 directly
//     into a rearranged LDS tensor R[18x18 halo][48 root ch][4 branches]:
//     the strided channel gather + space_to_depth + concat all happen inside
//     the async data mover (ASYNCcnt), no VGPR staging.
//   * R + transposed weights ~= 255KB LDS -> only possible with CDNA5's
//     320KB/WGP. 1 WG/WGP, 8 wave32s.
//   * Compute: 1 output px/thread, float4 accumulator over the 4 branches,
//     432 float4 FMAs from LDS (weights are LDS broadcasts), fused BN+ReLU.
//   * WMMA deliberately not used: each branch is a GEMV (N=1) over disjoint
//     data -> any 16x16x4 mapping wastes >=15/16 of the array and adds
//     instructions to a memory-bound kernel.
// ---------------------------------------------------------------------------

#define TILE      16
#define HALO      18                     // TILE + 2 (3x3 SAME halo)
#define PXS       196                    // floats/halo px: 192 data + 4 pad (16B aligned)
#define RFLOATS   (HALO * HALO * PXS)    // 63504 floats (~248 KB)
#define WRFLOATS  1728                   // 4*3*3*48 transposed weights
#define SMEM_FLOATS (WRFLOATS + RFLOATS)

__device__ __forceinline__ void async_cp16(unsigned lds_addr, const float* g) {
  // GLOBAL_LOAD_ASYNC_TO_LDS_B128 (GV mode): copies 16 bytes from the global address in g to the LDS address in lds_addr, counted by ASYNCcnt
  asm volatile("global_load_async_to_lds_b128 %0, %1, off"
               :: "v"(lds_addr), "v"(g)
               : "memory");
}

__device__ __forceinline__ void async_cp4(unsigned lds_addr, const float* g) {
  // GLOBAL_LOAD_ASYNC_TO_LDS_B32: single-dword async copy (weight transpose)
  asm volatile("global_load_async_to_lds_b32 %0, %1, off"
               :: "v"(lds_addr), "v"(g)
               : "memory");
}

__device__ __forceinline__ void wait_async0() {
#if __has_builtin(__builtin_amdgcn_s_wait_asynccnt)
  __builtin_amdgcn_s_wait_asynccnt(0);
#else
  asm volatile("s_wait_asynccnt 0x0" ::: "memory");
#endif
}

__global__ __launch_bounds__(256, 1)
void dense_next_blk_kernel(const float* __restrict__ x0,   // [32,256,256,64]
                           const float* __restrict__ x1,   // [32,128,128,96]
                           const float* __restrict__ x2,   // [32, 64, 64,128]
                           const float* __restrict__ cw,   // [4,3,3,48,1]
                           const float* __restrict__ gma,  // [4]
                           const float* __restrict__ bta,  // [4]
                           const float* __restrict__ mea,  // [4]
                           const float* __restrict__ vr,   // [4]
                           float* __restrict__ out)        // [4*32,64,64] flat
{
  extern __shared__ float smem[];
  float* Wr = smem;                 // Wr[(tap*48+c)*4 + i]
  float* R  = smem + WRFLOATS;      // R[px*PXS + c*4 + i]
  const unsigned Wb = (unsigned)(size_t)(Wr);  // low 32 bits = LDS byte offset
  const unsigned Rb = (unsigned)(size_t)(R);

  const int t  = threadIdx.x;
  const int bb = blockIdx.z;
  const int h0 = blockIdx.y * TILE;
  const int w0 = blockIdx.x * TILE;

  // ---- zero ONLY invalid halo pixels (SAME zero-padding), border tiles only
  const bool interior = (h0 >= 1) && (h0 + TILE + 1 <= 64) &&
                        (w0 >= 1) && (w0 + TILE + 1 <= 64);
  if (!interior) {
    float4 z = make_float4(0.f, 0.f, 0.f, 0.f);
    for (int px = t; px < 324; px += 256) {
      int hh = px / HALO, ww = px % HALO;
      int gh = h0 - 1 + hh, gw = w0 - 1 + ww;
      if ((unsigned)gh >= 64u || (unsigned)gw >= 64u) {
        float4* p = (float4*)(R + px * PXS);
        for (int q = 0; q < PXS / 4; ++q) p[q] = z;   // 49 b128 stores
      }
    }
    __syncthreads();   // zeros visible before anyone's async writes land
  }

  // ---- async weight transpose: Wr[(tap*48+c)*4+i] = cw[i*432 + tap*48 + c]
  for (int e = t; e < WRFLOATS; e += 256)
    async_cp4(Wb + (unsigned)(e * 4), cw + (e & 3) * 432 + (e >> 2));

  // ---- async gather into R -------------------------------------------------
  // Level 2 (x2): root ch 44+c'   chunk = x2[b, gh, gw, 32c' .. 32c'+3]
  for (int e = t; e < 324 * 4; e += 256) {
    int cp = e & 3, px = e >> 2;
    int hh = px / HALO, ww = px % HALO;
    int gh = h0 - 1 + hh, gw = w0 - 1 + ww;
    if ((unsigned)gh < 64u && (unsigned)gw < 64u) {
      const float* g = x2 + (((size_t)bb * 64 + gh) * 64 + gw) * 128 + cp * 32;
      async_cp16(Rb + (unsigned)((px * PXS + (44 + cp) * 4) * 4), g);
    }
  }
  // Level 1 (x1): block-2 s2d, root ch 32 + sub*3 + c'
  for (int e = t; e < 324 * 12; e += 256) {
    int cp = e % 3;  int r = e / 3;
    int sub = r & 3; int px = r >> 2;
    int hh = px / HALO, ww = px % HALO;
    int gh = h0 - 1 + hh, gw = w0 - 1 + ww;
    if ((unsigned)gh < 64u && (unsigned)gw < 64u) {
      int y = gh * 2 + (sub >> 1), x = gw * 2 + (sub & 1);
      const float* g = x1 + (((size_t)bb * 128 + y) * 128 + x) * 96 + cp * 32;
      async_cp16(Rb + (unsigned)((px * PXS + (32 + sub * 3 + cp) * 4) * 4), g);
    }
  }
  // Level 0 (x0): block-4 s2d, root ch sub*2 + c'
  for (int e = t; e < 324 * 32; e += 256) {
    int cp = e & 1;   int r = e >> 1;
    int sub = r & 15; int px = r >> 4;
    int hh = px / HALO, ww = px % HALO;
    int gh = h0 - 1 + hh, gw = w0 - 1 + ww;
    if ((unsigned)gh < 64u && (unsigned)gw < 64u) {
      int y = gh * 4 + (sub >> 2), x = gw * 4 + (sub & 3);
      const float* g = x0 + (((size_t)bb * 256 + y) * 256 + x) * 64 + cp * 32;
      async_cp16(Rb + (unsigned)((px * PXS + (sub * 2 + cp) * 4) * 4), g);
    }
  }

  wait_async0();     // ASYNCcnt == 0: my chunks (weights + data) are in LDS
  __syncthreads();   // everyone's chunks are in LDS

  // ---- compute: one output pixel per thread, 4 branches in a float4 -------
  const int ty = t >> 4, tx = t & 15;
  float ax = 0.f, ay = 0.f, az = 0.f, aw = 0.f;
  #pragma unroll
  for (int dy = 0; dy < 3; ++dy) {
    #pragma unroll
    for (int dx = 0; dx < 3; ++dx) {
      const float4* rp = (const float4*)(R  + ((ty + dy) * HALO + (tx + dx)) * PXS);
      const float4* wp = (const float4*)(Wr + (dy * 3 + dx) * 192);
      #pragma unroll 8
      for (int c = 0; c < 48; ++c) {
        float4 rv = rp[c], wv = wp[c];   // rv: per-lane b128, wv: LDS broadcast
        ax = fmaf(rv.x, wv.x, ax);
        ay = fmaf(rv.y, wv.y, ay);
        az = fmaf(rv.z, wv.z, az);
        aw = fmaf(rv.w, wv.w, aw);
      }
    }
  }

  const int h = h0 + ty, w = w0 + tx;
  float acc[4] = {ax, ay, az, aw};
  #pragma unroll
  for (int i = 0; i < 4; ++i) {
    float sc = gma[i] * rsqrtf(vr[i] + 1e-5f);
    float v  = (acc[i] - mea[i]) * sc + bta[i];
    out[((size_t)i * 32 + bb) * 4096 + (size_t)h * 64 + w] = fmaxf(v, 0.f);
  }
}

extern "C" void kernel_launch(void* const* d_in, const int* in_sizes, int n_in,
                              void* d_out, int out_size, void* d_ws, size_t ws_size,
                              hipStream_t stream) {
  (void)in_sizes; (void)n_in; (void)out_size; (void)d_ws; (void)ws_size;
  const float* x0  = (const float*)d_in[0];
  const float* x1  = (const float*)d_in[1];
  const float* x2  = (const float*)d_in[2];
  const float* cw  = (const float*)d_in[3];
  const float* gma = (const float*)d_in[4];
  const float* bta = (const float*)d_in[5];
  const float* mea = (const float*)d_in[6];
  const float* vr  = (const float*)d_in[7];
  float* out = (float*)d_out;

  dim3 grid(64 / TILE, 64 / TILE, 32);   // 4 x 4 tiles x 32 batches = 512 WGs
  dim3 block(256);                       // 8 wave32s, 1 WG per WGP (~255KB LDS)
  size_t smem_bytes = (size_t)SMEM_FLOATS * sizeof(float);  // < 320 KB/WGP

  hipLaunchKernelGGL(dense_next_blk_kernel, grid, block, smem_bytes, stream,
                     x0, x1, x2, cw, gma, bta, mea, vr, out);
}